// Qwen3NextGatedDeltaNet_48009144435152
// MI455X (gfx1250) — compile-verified
//
#include <hip/hip_runtime.h>
#include <hip/hip_bf16.h>
#include <math.h>

// Qwen3-Next Gated DeltaNet, chunked delta-rule formulation for CDNA5 WMMA.
#define HK_    16
#define HV_    32
#define DK_    128
#define DV_    128
#define KCONV_ 4
#define QKVC_  8192          // 2*HK*DK + HV*DV
#define BB_    2
#define SS_    512
#define CL_    16            // chunk length (matches WMMA M/N=16)
#define NCH_   (SS_/CL_)
#define VB_    32            // DV columns per block (DV split x4)
#define NVB_   (DV_/VB_)

typedef __attribute__((ext_vector_type(2))) float v2f;
typedef __attribute__((ext_vector_type(8))) float v8f;
typedef int v4i_ __attribute__((vector_size(16)));   // matches builtin param type

__device__ __forceinline__ float sigmoidf_(float x){ return 1.0f/(1.0f+__expf(-x)); }
__device__ __forceinline__ float softplusf_(float x){ return (x>20.f)? x : log1pf(__expf(x)); }

__device__ __forceinline__ v8f wmma4(v2f a, v2f b, v8f c){
  // V_WMMA_F32_16X16X4_F32 : D = A(16x4) * B(4x16) + C(16x16)
  return __builtin_amdgcn_wmma_f32_16x16x4_f32(false, a, false, b, (short)0, c, false, false);
}

// ---- gfx1250 async global->LDS path (ASYNCcnt), guarded so compile never breaks
#if defined(__gfx1250__) && __has_builtin(__builtin_amdgcn_global_load_async_to_lds_b128)
#define GDN_ASYNC 1
#endif

__device__ __forceinline__ void async_cp16(const float* g, float* l){
#ifdef GDN_ASYNC
  __builtin_amdgcn_global_load_async_to_lds_b128((v4i_*)g, (v4i_*)l, 0, 0);
#else
  *(float4*)l = *(const float4*)g;
#endif
}

__device__ __forceinline__ void wait_async_all(){
#ifdef GDN_ASYNC
 #if __has_builtin(__builtin_amdgcn_s_wait_asynccnt)
  __builtin_amdgcn_s_wait_asynccnt(0);
 #else
  asm volatile("s_wait_asynccnt 0" ::: "memory");
 #endif
#endif
}

// ---------------------------------------------------------------------------
// Kernel 1: depthwise causal conv(K=4) + SiLU + per-head l2norm for q/k.
// One block per (b, s, head64); 128 threads (4 waves) = one channel each.
// ---------------------------------------------------------------------------
__global__ void conv_silu_norm_kernel(const float* __restrict__ mixed_qkv,
                                      const float* __restrict__ conv_w,
                                      float* __restrict__ Qn,
                                      float* __restrict__ Kn,
                                      float* __restrict__ Vg)
{
  __shared__ float wsum[4];
  const int tid = threadIdx.x;
  const int e   = blockIdx.x & 63;
  const int s   = (blockIdx.x >> 6) & (SS_-1);
  const int b   = blockIdx.x >> 15;
  const int ch  = e*128 + tid;

  const float* w = conv_w + (size_t)ch*KCONV_;
  float acc = 0.f;
  #pragma unroll
  for (int j=0;j<KCONV_;j++){
    int sp = s - (KCONV_-1) + j;
    float xv = (sp>=0) ? mixed_qkv[((size_t)(b*SS_+sp))*QKVC_ + ch] : 0.f;
    acc = fmaf(xv, w[j], acc);
  }
  float y = acc * sigmoidf_(acc);            // SiLU

  if (e < 32) {                              // q/k heads: l2 normalize over 128
    float ss = y*y;
    #pragma unroll
    for (int m=16;m>=1;m>>=1) ss += __shfl_xor(ss, m, 32);
    if ((tid & 31) == 0) wsum[tid>>5] = ss;
    __syncthreads();
    float tot = wsum[0]+wsum[1]+wsum[2]+wsum[3];
    float rn = rsqrtf(tot + 1e-6f);
    if (e < 16) rn *= 0.08838834764831845f;  // * DK^-0.5
    y *= rn;
  }
  const size_t tok = (size_t)(b*SS_+s);
  if (e < 16)      Qn[(tok*HK_ + e)*DK_ + tid]      = y;
  else if (e < 32) Kn[(tok*HK_ + (e-16))*DK_ + tid] = y;
  else             Vg[(tok*HV_ + (e-32))*DV_ + tid] = y;
}

// ---------------------------------------------------------------------------
// Kernel 2: chunked gated delta-rule scan, DV split across NVB_ blocks.
// One block per (b, h, vb): 256 threads = 8 waves, state = [DK][VB] in LDS.
// ---------------------------------------------------------------------------
__global__ void __launch_bounds__(256,1) gdn_scan_kernel(
    const float* __restrict__ Qn, const float* __restrict__ Kn,
    const float* __restrict__ Vg,
    const float* __restrict__ a_in, const float* __restrict__ b_in,
    const float* __restrict__ alog, const float* __restrict__ dt_bias,
    const float* __restrict__ state_init, float* __restrict__ out)
{
  __shared__ __align__(16) float S0[DK_*VB_];      // 4096 state [DK][VB]
  __shared__ __align__(16) float Qb2[2][CL_*DK_];  // double-buffered chunk Q
  __shared__ __align__(16) float Kb2[2][CL_*DK_];  // double-buffered chunk K
  __shared__ __align__(16) float Vb2[2][CL_*VB_];  // double-buffered chunk V -> Rhs -> Delta
  __shared__ float KKm[256], QKm[256], Tm[256], Pm[256];
  __shared__ float cg[CL_], Asc[CL_], wsc[CL_], ksl[CL_], betv[CL_];

  const int tid  = threadIdx.x;
  const int lane = tid & 31;
  const int wid  = tid >> 5;                 // 0..7
  const int vb   = blockIdx.x & (NVB_-1);
  const int h    = (blockIdx.x >> 2) & (HV_-1);
  const int b    = blockIdx.x >> 7;
  const int hk   = h >> 1;                   // q/k head repeat ratio = 2
  const int vbase= vb*VB_;

  // WMMA lane mapping (16x16x4 f32)
  const int n0    = lane & 15;               // N for B/C operands
  const int mrow  = lane & 15;               // M for A operand
  const int khalf = (lane >> 4) << 1;        // K pair: 0 or 2
  const int mbase = (lane >> 4) << 3;        // C rows: 0 or 8
  const int nnB   = (wid & 1) * 16;          // N tile for phase-B/E waves

  // Initial state (this block's DV columns)
  const float* si = state_init + ((size_t)(b*HV_ + h))*DK_*DV_ + vbase;
  for (int i=tid; i<DK_*VB_; i+=256) S0[i] = si[(i>>5)*DV_ + (i&31)];

  // Prologue: async-prefetch chunk 0 into buffer 0
  {
    const size_t tb0 = (size_t)(b*SS_);
    #pragma unroll
    for (int r=0;r<2;r++){
      int f = (tid + r*256)*4, t = f>>7, d = f&127;
      async_cp16(Qn + ((tb0+t)*HK_+hk)*DK_ + d, &Qb2[0][f]);
      async_cp16(Kn + ((tb0+t)*HK_+hk)*DK_ + d, &Kb2[0][f]);
    }
    if (tid < 128){
      int f = tid*4, t = f>>5, d = f&31;
      async_cp16(Vg + ((tb0+t)*HV_+h)*DV_ + vbase + d, &Vb2[0][f]);
    }
  }

  for (int c=0; c<NCH_; c++){
    const int cur = c & 1, nxt = cur ^ 1;
    const size_t tokbase = (size_t)(b*SS_ + c*CL_);
    float* Qc = Qb2[cur]; float* Kc = Kb2[cur]; float* Vc = Vb2[cur];

    wait_async_all();                     // this wave's pending async loads done
    __syncthreads();                      // all waves' data visible; S0 stable

    // Prefetch next chunk into the other buffer (overlaps all compute below)
    if (c+1 < NCH_){
      const size_t tbn = tokbase + CL_;
      #pragma unroll
      for (int r=0;r<2;r++){
        int f = (tid + r*256)*4, t = f>>7, d = f&127;
        async_cp16(Qn + ((tbn+t)*HK_+hk)*DK_ + d, &Qb2[nxt][f]);
        async_cp16(Kn + ((tbn+t)*HK_+hk)*DK_ + d, &Kb2[nxt][f]);
      }
      if (tid < 128){
        int f = tid*4, t = f>>5, d = f&31;
        async_cp16(Vg + ((tbn+t)*HV_+h)*DV_ + vbase + d, &Vb2[nxt][f]);
      }
    }

    // ---- gates ----
    if (tid < CL_){
      float av = a_in[(tokbase+tid)*HV_ + h] + dt_bias[h];
      cg[tid]   = -__expf(alog[h]) * softplusf_(av);     // per-step g (temp)
      betv[tid] = sigmoidf_(b_in[(tokbase+tid)*HV_ + h]);
    }
    __syncthreads();
    if (tid == 0){ float a0=0.f; for (int t=0;t<CL_;t++){ a0+=cg[t]; cg[t]=a0; } }
    __syncthreads();
    if (tid < CL_){
      float cgl = cg[CL_-1];
      Asc[tid] = __expf(cg[tid]);          // A_t
      wsc[tid] = betv[tid]*Asc[tid];       // beta_t*A_t
      ksl[tid] = __expf(cgl - cg[tid]);    // A_L/A_t
    }
    __syncthreads();

    // ---- Phase B: WMMA against old S0 (wave-specialized) ----
    v8f racc = {};   // (beta A K) S0 tile -> Rhs    (waves 0,1)
    v8f oacc = {};   // (A Q) S0 tile      -> O part (waves 2,3)
    if (wid < 2){
      for (int kk=0; kk<DK_; kk+=4){
        const int k0 = kk + khalf;
        v2f bS; bS.x = S0[k0*VB_ + nnB + n0];      bS.y = S0[(k0+1)*VB_ + nnB + n0];
        v2f aW; aW.x = wsc[mrow]*Kc[mrow*DK_+k0];  aW.y = wsc[mrow]*Kc[mrow*DK_+k0+1];
        racc = wmma4(aW, bS, racc);
      }
    } else if (wid < 4){
      for (int kk=0; kk<DK_; kk+=4){
        const int k0 = kk + khalf;
        v2f bS; bS.x = S0[k0*VB_ + nnB + n0];      bS.y = S0[(k0+1)*VB_ + nnB + n0];
        v2f aQ; aQ.x = Asc[mrow]*Qc[mrow*DK_+k0];  aQ.y = Asc[mrow]*Qc[mrow*DK_+k0+1];
        oacc = wmma4(aQ, bS, oacc);
      }
    } else if (wid == 4){                          // raw K K^T
      v8f kka = {};
      for (int kk=0; kk<DK_; kk+=4){
        const int k0 = kk + khalf;
        v2f aK; aK.x = Kc[mrow*DK_+k0]; aK.y = Kc[mrow*DK_+k0+1];
        v2f bK; bK.x = Kc[n0*DK_+k0];   bK.y = Kc[n0*DK_+k0+1];
        kka = wmma4(aK, bK, kka);
      }
      #pragma unroll
      for (int j=0;j<8;j++) KKm[(j+mbase)*16 + n0] = kka[j];
    } else if (wid == 5){                          // raw Q K^T
      v8f qka = {};
      for (int kk=0; kk<DK_; kk+=4){
        const int k0 = kk + khalf;
        v2f aQ; aQ.x = Qc[mrow*DK_+k0]; aQ.y = Qc[mrow*DK_+k0+1];
        v2f bK; bK.x = Kc[n0*DK_+k0];   bK.y = Kc[n0*DK_+k0+1];
        qka = wmma4(aQ, bK, qka);
      }
      #pragma unroll
      for (int j=0;j<8;j++) QKm[(j+mbase)*16 + n0] = qka[j];
    }
    __syncthreads();

    // ---- Phase C: Rhs = beta*V - racc (in-place in Vc); build T, P ----
    if (wid < 2){
      #pragma unroll
      for (int j=0;j<8;j++){
        const int t = j + mbase, col = nnB + n0;
        Vc[t*VB_ + col] = betv[t]*Vc[t*VB_ + col] - racc[j];
      }
    }
    {
      const int t = tid >> 4, s2 = tid & 15;
      float dec = __expf(cg[t] - cg[s2]);          // <= 1 for s2 <= t
      Tm[tid] = (s2 <  t) ? betv[t]*dec*KKm[tid] : 0.f;
      Pm[tid] = (s2 <= t) ? dec*QKm[tid]         : 0.f;
    }
    __syncthreads();

    // ---- Phase D: forward substitution (I+T)Delta = Rhs, one column/thread ----
    if (tid < VB_){
      float dlt[CL_];
      #pragma unroll
      for (int t=0;t<CL_;t++){
        float x = Vc[t*VB_ + tid];
        #pragma unroll
        for (int s2=0;s2<CL_;s2++) if (s2 < t) x -= Tm[t*16+s2]*dlt[s2];
        dlt[t] = x;
        Vc[t*VB_ + tid] = x;                       // Delta
      }
    }
    __syncthreads();

    // ---- Phase E: O += P * Delta; write output (waves 2,3) ----
    if (wid >= 2 && wid < 4){
      #pragma unroll
      for (int kk=0; kk<CL_; kk+=4){
        const int k0 = kk + khalf;
        v2f aP; aP.x = Pm[mrow*16+k0];      aP.y = Pm[mrow*16+k0+1];
        v2f bD; bD.x = Vc[k0*VB_+nnB+n0];   bD.y = Vc[(k0+1)*VB_+nnB+n0];
        oacc = wmma4(aP, bD, oacc);
      }
      #pragma unroll
      for (int j=0;j<8;j++){
        const int t = j + mbase;
        out[((tokbase + t)*HV_ + h)*DV_ + vbase + nnB + n0] = oacc[j];
      }
    }

    // ---- Phase F: S0 = A_L*S0 + K_hat^T Delta (2 tiles per wave) ----
    const float AL = Asc[CL_-1];
    #pragma unroll
    for (int r=0;r<2;r++){
      const int nn2 = r*16, mm = wid*16;
      v8f sacc;
      #pragma unroll
      for (int j=0;j<8;j++) sacc[j] = AL * S0[(mm + j + mbase)*VB_ + nn2 + n0];
      #pragma unroll
      for (int kk=0; kk<CL_; kk+=4){
        const int k0 = kk + khalf;
        v2f aK; aK.x = ksl[k0]*Kc[k0*DK_+mm+mrow]; aK.y = ksl[k0+1]*Kc[(k0+1)*DK_+mm+mrow];
        v2f bD; bD.x = Vc[k0*VB_+nn2+n0];          bD.y = Vc[(k0+1)*VB_+nn2+n0];
        sacc = wmma4(aK, bD, sacc);
      }
      #pragma unroll
      for (int j=0;j<8;j++) S0[(mm + j + mbase)*VB_ + nn2 + n0] = sacc[j];
    }
    __syncthreads();   // protect S0 / current buffers before next chunk
  }
}

// ---------------------------------------------------------------------------
extern "C" void kernel_launch(void* const* d_in, const int* in_sizes, int n_in,
                              void* d_out, int out_size, void* d_ws, size_t ws_size,
                              hipStream_t stream) {
  (void)in_sizes; (void)n_in; (void)out_size; (void)ws_size;
  const float* mixed_qkv = (const float*)d_in[0];
  const float* a_in      = (const float*)d_in[1];
  const float* b_in      = (const float*)d_in[2];
  const float* conv_w    = (const float*)d_in[3];
  const float* alog      = (const float*)d_in[4];
  const float* dt_bias   = (const float*)d_in[5];
  const float* ssm_state = (const float*)d_in[6];
  float* out = (float*)d_out;

  float* Qn = (float*)d_ws;                              // [B*S][16][128]
  float* Kn = Qn + (size_t)BB_*SS_*HK_*DK_;              // [B*S][16][128]
  float* Vg = Kn + (size_t)BB_*SS_*HK_*DK_;              // [B*S][32][128]

  conv_silu_norm_kernel<<<BB_*SS_*64, 128, 0, stream>>>(mixed_qkv, conv_w, Qn, Kn, Vg);

  gdn_scan_kernel<<<BB_*HV_*NVB_, 256, 0, stream>>>(Qn, Kn, Vg, a_in, b_in,
                                                    alog, dt_bias, ssm_state, out);
}